// HAN1Layer_80547816669336
// MI455X (gfx1250) — compile-verified
//
#include <hip/hip_runtime.h>
#include <math.h>

#define NN 200000
#define EE 3000000
#define FIN 64
#define HH 8

typedef __attribute__((ext_vector_type(2))) float v2f;
typedef __attribute__((ext_vector_type(8))) float v8f;

// ---------- monotone float<->uint key (for deterministic atomic max) ----------
__device__ __forceinline__ unsigned fkey(float f) {
  unsigned u = __float_as_uint(f);
  return (u & 0x80000000u) ? ~u : (u | 0x80000000u);
}
__device__ __forceinline__ float funkey(unsigned k) {
  unsigned u = (k & 0x80000000u) ? (k ^ 0x80000000u) : ~k;
  return __uint_as_float(u);
}
#define NEG_INF_KEY 0x007FFFFFu  // fkey(-inf)

// ---------- projection: h = x @ W + b via V_WMMA_F32_16X16X4_F32 ----------
// one wave -> 16 rows x 16 cols (8 valid) tile; K=64 via 16 k-steps of 4.
// W is staged in LDS, zero-padded to 16 columns, K-pair interleaved so each
// lane's B fragment is one unconditional ds_load_b64 (no divergent branches).
__global__ void proj_wmma(const float* __restrict__ x,
                          const float* __restrict__ W,   // [64,8] row-major
                          const float* __restrict__ b,   // [8]
                          float* __restrict__ hout) {    // [N,8]
  // Wp layout: pair p = k/2 (0..31), column n (0..15), kq = k&1:
  //   Wp[p*32 + n*2 + kq] = (n<8) ? W[(2p+kq)*8 + n] : 0
  __shared__ float Wp[64 * 16];
  __shared__ float bias_s[16];

  for (int i = threadIdx.x; i < 64 * 16; i += blockDim.x) {
    int p  = i >> 5;
    int r  = i & 31;
    int n  = r >> 1;
    int kq = r & 1;
    float v = W[(size_t)(2 * p + kq) * HH + (n & 7)];  // always in-bounds
    Wp[i] = (n < HH) ? v : 0.0f;                       // cndmask, no branch
  }
  if (threadIdx.x < 16) {
    float v = b[threadIdx.x & 7];
    bias_s[threadIdx.x] = (threadIdx.x < HH) ? v : 0.0f;
  }
  __syncthreads();

  int lane = threadIdx.x & 31;
  int wave = threadIdx.x >> 5;
  int tile = blockIdx.x * 4 + wave;
  int row0 = tile * 16;
  int l  = lane & 15;   // M (for A) / N (for B,C)
  int hi = lane >> 4;   // selects upper K-pair / upper M-half

  float bias = bias_s[l];
  v8f c;
#pragma unroll
  for (int r = 0; r < 8; ++r) c[r] = bias;

  const float* arow = x + (size_t)(row0 + l) * FIN + hi * 2;
  const float* brow = Wp + hi * 32 + l * 2;
#pragma unroll
  for (int kk = 0; kk < 16; ++kk) {
    v2f a, bb;
    a.x = arow[kk * 4 + 0];            // A[M=l][K=4kk+2hi]
    a.y = arow[kk * 4 + 1];            // A[M=l][K=4kk+2hi+1]
    bb.x = brow[kk * 64 + 0];          // B[K=4kk+2hi  ][N=l]
    bb.y = brow[kk * 64 + 1];          // B[K=4kk+2hi+1][N=l]
    c = __builtin_amdgcn_wmma_f32_16x16x4_f32(false, a, false, bb,
                                              (short)0, c, false, false);
  }
  if (l < HH) {
    float* hrow = hout + (size_t)row0 * HH + l;
#pragma unroll
    for (int r = 0; r < 8; ++r)
      hrow[(size_t)(r + 8 * hi) * HH] = c[r];  // D[M=r+8hi][N=l]
  }
}

// ---------- per-node attention scalars: 8 arrays of N ----------
__device__ __forceinline__ float dot8(const float* __restrict__ h,
                                      const float* __restrict__ a) {
  float s = 0.f;
#pragma unroll
  for (int i = 0; i < 8; ++i) s += h[i] * a[i];
  return s;
}

__global__ void alpha_kernel(const float* __restrict__ h_ind,
                             const float* __restrict__ h_org,
                             const float* __restrict__ h_ext,
                             const float* as_io, const float* ad_io,
                             const float* as_oi, const float* ad_oi,
                             const float* as_ei, const float* ad_ei,
                             const float* as_eo, const float* ad_eo,
                             float* __restrict__ alpha) {
  int n = blockIdx.x * blockDim.x + threadIdx.x;
  if (n >= NN) return;
  const float* hi = h_ind + (size_t)n * 8;
  const float* ho = h_org + (size_t)n * 8;
  const float* he = h_ext + (size_t)n * 8;
  alpha[0 * NN + n] = dot8(hi, as_io);
  alpha[1 * NN + n] = dot8(ho, ad_io);
  alpha[2 * NN + n] = dot8(ho, as_oi);
  alpha[3 * NN + n] = dot8(hi, ad_oi);
  alpha[4 * NN + n] = dot8(he, as_ei);
  alpha[5 * NN + n] = dot8(hi, ad_ei);
  alpha[6 * NN + n] = dot8(he, as_eo);
  alpha[7 * NN + n] = dot8(ho, ad_eo);
}

// ---------- workspace init ----------
__global__ void fill_u32(unsigned* __restrict__ p, unsigned val, long count) {
  long i = (long)blockIdx.x * blockDim.x + threadIdx.x;
  if (i < count) p[i] = val;
}

// ---------- GAT pass A: per-segment max via atomic max on monotone keys ----------
__global__ void gat_max(const int* __restrict__ rows, const int* __restrict__ cols,
                        const float* __restrict__ as, const float* __restrict__ ad,
                        unsigned* __restrict__ mkeys) {
  int e = blockIdx.x * blockDim.x + threadIdx.x;
  if (e >= EE) return;
  int r = rows[e], c = cols[e];
  float a = as[r] + ad[c];
  a = (a > 0.f) ? a : 0.2f * a;  // leaky_relu 0.2
  atomicMax(mkeys + c, fkey(a));
}

// ---------- GAT pass B: exp weights, denominator, weighted scatter-add ----------
// 8 lanes per edge (lane = feature) -> coalesced h_src gather & accumulate
__global__ void gat_acc(const int* __restrict__ rows, const int* __restrict__ cols,
                        const float* __restrict__ as, const float* __restrict__ ad,
                        const unsigned* __restrict__ mkeys,
                        const float* __restrict__ hsrc,
                        float* __restrict__ den, float* __restrict__ acc) {
  long t = (long)blockIdx.x * blockDim.x + threadIdx.x;
  long e = t >> 3;
  int  f = (int)(t & 7);
  if (e >= EE) return;
  int r = rows[e], c = cols[e];
  float a = as[r] + ad[c];
  a = (a > 0.f) ? a : 0.2f * a;
  float m  = funkey(mkeys[c]);
  float ex = expf(a - m);
  if (f == 0) atomicAdd(den + c, ex);
  atomicAdd(acc + (size_t)c * 8 + f, hsrc[(size_t)r * 8 + f] * ex);
}

// ---------- normalize + relu, all 4 accumulators at once ----------
__global__ void norm_relu(float* __restrict__ acc, const float* __restrict__ den) {
  long i = (long)blockIdx.x * blockDim.x + threadIdx.x;  // over 4*N*8
  if (i >= 4L * NN * 8) return;
  float d = den[i >> 3] + 1e-16f;     // (i>>3) == t*N + n
  float v = acc[i] / d;
  acc[i] = (v > 0.f) ? v : 0.f;
}

// ---------- semantic attention score: sum over nodes of tanh(o kW + kb).q ----------
__global__ void sem_score(const float* __restrict__ acc,
                          const float* __restrict__ kW, const float* __restrict__ kb,
                          const float* __restrict__ q, float* __restrict__ scores) {
  int a = blockIdx.y;
  const float* o = acc + (size_t)a * NN * 8;
  int n = blockIdx.x * blockDim.x + threadIdx.x;
  float s = 0.f;
  if (n < NN) {
    const float* on = o + (size_t)n * 8;
#pragma unroll
    for (int j = 0; j < 8; ++j) {
      float t = kb[j];
#pragma unroll
      for (int i = 0; i < 8; ++i) t += on[i] * kW[i * 8 + j];
      s += tanhf(t) * q[j];
    }
  }
  __shared__ float sd[256];
  sd[threadIdx.x] = s;
  __syncthreads();
  for (int off = 128; off > 0; off >>= 1) {
    if (threadIdx.x < off) sd[threadIdx.x] += sd[threadIdx.x + off];
    __syncthreads();
  }
  if (threadIdx.x == 0) atomicAdd(scores + a, sd[0]);
}

// ---------- semantic softmax combine + sigmoid head ----------
__global__ void final_pred(const float* __restrict__ o0, const float* __restrict__ o1,
                           const float* __restrict__ sc2,  // two score sums
                           const float* __restrict__ linW, const float* __restrict__ linb,
                           float* __restrict__ out) {
  int n = blockIdx.x * blockDim.x + threadIdx.x;
  if (n >= NN) return;
  float s0 = sc2[0] * (1.0f / NN);
  float s1 = sc2[1] * (1.0f / NN);
  float mx = fmaxf(s0, s1);
  float e0 = expf(s0 - mx), e1 = expf(s1 - mx);
  float inv = 1.0f / (e0 + e1);
  float w0 = e0 * inv, w1 = e1 * inv;
  float z = linb[0];
#pragma unroll
  for (int h = 0; h < 8; ++h)
    z += (w0 * o0[(size_t)n * 8 + h] + w1 * o1[(size_t)n * 8 + h]) * linW[h];
  out[n] = 1.0f / (1.0f + expf(-z));
}

extern "C" void kernel_launch(void* const* d_in, const int* in_sizes, int n_in,
                              void* d_out, int out_size, void* d_ws, size_t ws_size,
                              hipStream_t stream) {
  const float* x_ind = (const float*)d_in[0];
  const float* x_org = (const float*)d_in[1];
  const float* x_ext = (const float*)d_in[2];
  const int* ei_io = (const int*)d_in[3];
  const int* ei_oi = (const int*)d_in[4];
  const int* ei_ei = (const int*)d_in[5];
  const int* ei_eo = (const int*)d_in[6];
  const float* W_ind = (const float*)d_in[7];  const float* b_ind = (const float*)d_in[8];
  const float* W_org = (const float*)d_in[9];  const float* b_org = (const float*)d_in[10];
  const float* W_ext = (const float*)d_in[11]; const float* b_ext = (const float*)d_in[12];
  const float* as_io = (const float*)d_in[13]; const float* ad_io = (const float*)d_in[14];
  const float* as_oi = (const float*)d_in[15]; const float* ad_oi = (const float*)d_in[16];
  const float* as_ei = (const float*)d_in[17]; const float* ad_ei = (const float*)d_in[18];
  const float* as_eo = (const float*)d_in[19]; const float* ad_eo = (const float*)d_in[20];
  const float* kW = (const float*)d_in[21];
  const float* kb = (const float*)d_in[22];
  const float* q  = (const float*)d_in[23];
  const float* lin_ind_W = (const float*)d_in[24]; const float* lin_ind_b = (const float*)d_in[25];
  const float* lin_org_W = (const float*)d_in[26]; const float* lin_org_b = (const float*)d_in[27];
  float* out = (float*)d_out;

  // workspace layout (floats)
  float* ws = (float*)d_ws;
  float*   h_ind = ws;                       // 8N
  float*   h_org = h_ind + (size_t)8 * NN;   // 8N
  float*   h_ext = h_org + (size_t)8 * NN;   // 8N
  float*   alpha = h_ext + (size_t)8 * NN;   // 8N  (8 arrays of N)
  unsigned* mkeys = (unsigned*)(alpha + (size_t)8 * NN);  // 4N
  float*   den   = (float*)(mkeys + (size_t)4 * NN);      // 4N
  float*   acc   = den + (size_t)4 * NN;                  // 32N (4 arrays of 8N)
  float*   scores = acc + (size_t)32 * NN;                // 16

  // 1) projections via WMMA (fp32): 12500 tiles of 16 rows, 4 waves/block
  dim3 pb(128), pg(NN / 16 / 4);
  proj_wmma<<<pg, pb, 0, stream>>>(x_ind, W_ind, b_ind, h_ind);
  proj_wmma<<<pg, pb, 0, stream>>>(x_org, W_org, b_org, h_org);
  proj_wmma<<<pg, pb, 0, stream>>>(x_ext, W_ext, b_ext, h_ext);

  // 2) per-node alpha scalars
  alpha_kernel<<<(NN + 255) / 256, 256, 0, stream>>>(
      h_ind, h_org, h_ext, as_io, ad_io, as_oi, ad_oi, as_ei, ad_ei, as_eo, ad_eo, alpha);

  // 3) init max-keys to -inf key; zero den/acc/scores (contiguous)
  {
    long cnt = 4L * NN;
    fill_u32<<<(int)((cnt + 255) / 256), 256, 0, stream>>>(mkeys, NEG_INF_KEY, cnt);
    long zcnt = 36L * NN + 16;
    fill_u32<<<(int)((zcnt + 255) / 256), 256, 0, stream>>>((unsigned*)den, 0u, zcnt);
  }

  // 4) GAT passes; slot order: 0=org->ind 1=ext->ind 2=ind->org 3=ext->org
  const int* rowsArr[4] = {ei_oi, ei_ei, ei_io, ei_eo};
  const float* asArr[4] = {alpha + 2L * NN, alpha + 4L * NN, alpha + 0L * NN, alpha + 6L * NN};
  const float* adArr[4] = {alpha + 3L * NN, alpha + 5L * NN, alpha + 1L * NN, alpha + 7L * NN};
  const float* hsArr[4] = {h_org, h_ext, h_ind, h_ext};
  int gA = (EE + 255) / 256;
  int gB = (int)((8L * EE + 255) / 256);
  for (int t = 0; t < 4; ++t) {
    const int* rows = rowsArr[t];
    const int* cols = rows + EE;
    gat_max<<<gA, 256, 0, stream>>>(rows, cols, asArr[t], adArr[t], mkeys + (size_t)t * NN);
    gat_acc<<<gB, 256, 0, stream>>>(rows, cols, asArr[t], adArr[t],
                                    mkeys + (size_t)t * NN, hsArr[t],
                                    den + (size_t)t * NN, acc + (size_t)t * 8 * NN);
  }

  // 5) normalize + relu (all 4 metapath outputs)
  {
    long cnt = 4L * NN * 8;
    norm_relu<<<(int)((cnt + 255) / 256), 256, 0, stream>>>(acc, den);
  }

  // 6) semantic scores (4 metapath outputs -> 4 scalars)
  {
    dim3 g((NN + 255) / 256, 4);
    sem_score<<<g, 256, 0, stream>>>(acc, kW, kb, q, scores);
  }

  // 7) softmax over metapaths + prediction heads
  int gF = (NN + 255) / 256;
  final_pred<<<gF, 256, 0, stream>>>(acc + 0L * 8 * NN, acc + 1L * 8 * NN,
                                     scores + 0, lin_ind_W, lin_ind_b, out);
  final_pred<<<gF, 256, 0, stream>>>(acc + 2L * 8 * NN, acc + 3L * 8 * NN,
                                     scores + 2, lin_org_W, lin_org_b, out + NN);
}